// WeightedAverage_color_25692494365438
// MI455X (gfx1250) — compile-verified
//
#include <hip/hip_runtime.h>
#include <cstdint>

// ---------------------------------------------------------------------------
// WeightedAverage_color: fused 3x3 neighborhood color-affinity softmax.
//   l = x_lab[:,0]*50+50 ; a = x_lab[:,1] ; b = x_lab[:,2]
//   diff_k = (l_k-l)^2 + (a_k-a)^2 + (b_k-b)^2   (9 neighbors, zero-padded)
//   corr   = softmax_k(-diff)    [center logit == 0 -> max-free softmax]
//   out    = [sum corr*a_pred_k , sum corr*b_pred_k]
// Memory-bound (~59 MB traffic, ~0.25 GFLOP -> ~2.5us at 23.3 TB/s).
// Staging: TDM tensor_load_to_lds (5 descriptors/block) for interior blocks;
// per-lane global_load_async_to_lds_b32 fallback for top/left border blocks.
// ---------------------------------------------------------------------------

#define HH   512
#define WW   512
#define NPLANE (HH * WW)
#define TW   64                  // tile width  (== blockDim.x)
#define TH   16                  // tile height (4 rows per thread, blockDim.y=4)
#define HTW  (TW + 2)            // 66 halo width
#define HTH  (TH + 2)            // 18 halo height
#define HTILE (HTW * HTH)        // 1188 elements per channel plane in LDS

#if defined(__has_builtin)
#  if __has_builtin(__builtin_amdgcn_tensor_load_to_lds)
#    define HAVE_TDM 1
#  else
#    define HAVE_TDM 0
#  endif
#else
#  define HAVE_TDM 0
#endif

typedef unsigned int v4u __attribute__((ext_vector_type(4)));
typedef int          v8i __attribute__((ext_vector_type(8)));
typedef int          v4i __attribute__((ext_vector_type(4)));

// CDNA5 async copy: global memory -> LDS, no VGPR staging, tracked by ASYNCcnt.
__device__ __forceinline__ void async_f32_to_lds(uint32_t lds_byte, const float* g) {
    asm volatile("global_load_async_to_lds_b32 %0, %1, off"
                 :: "v"(lds_byte), "v"(g)
                 : "memory");
}

#if HAVE_TDM
// One TDM descriptor: copy a HTW x HTH f32 tile starting at (gy0,gx0) of a
// 512x512 plane into LDS (rows packed at HTW stride). tensor_dim* are the
// remaining extents from the tile origin, so right/bottom OOB reads return 0
// (exactly the zero padding the reference needs).
__device__ __forceinline__ void tdm_load_tile(const float* gbase, uint32_t lds_byte,
                                              int gx0, int gy0) {
    const uint64_t gaddr = (uint64_t)(uintptr_t)(gbase + (size_t)gy0 * WW + gx0);
    const uint32_t td0 = (uint32_t)(WW - gx0);   // remaining cols from tile start
    const uint32_t td1 = (uint32_t)(HH - gy0);   // remaining rows from tile start

    v4u g0;
    g0.x = 1u;                                            // count=1, flags=0
    g0.y = lds_byte;                                      // lds_addr (bytes)
    g0.z = (uint32_t)gaddr;                               // global_addr[31:0]
    g0.w = ((uint32_t)(gaddr >> 32) & 0x01FFFFFFu)        // global_addr[56:32]
         | (2u << 30);                                    // type = 2 ("image")

    v8i g1;
    g1[0] = (int)(2u << 16);                              // data_size=2 (4B), mask=0
    g1[1] = (int)((td0 & 0xFFFFu) << 16);                 // tensor_dim0[15:0]
    g1[2] = (int)(((td0 >> 16) & 0xFFFFu)                 // tensor_dim0[31:16]
         |  ((td1 & 0xFFFFu) << 16));                     // tensor_dim1[15:0]
    g1[3] = (int)(((td1 >> 16) & 0xFFFFu)                 // tensor_dim1[31:16]
         |  ((uint32_t)HTW << 16));                       // tile_dim0 = 66
    g1[4] = (int)HTH;                                     // tile_dim1=18, tile_dim2=0
    g1[5] = (int)WW;                                      // tensor_dim0_stride = 512
    g1[6] = 0;                                            // stride0[47:32], stride1 lo
    g1[7] = 0;                                            // stride1 hi

    const v4i g2 = {0, 0, 0, 0};                          // 2D: groups 2/3 unused
    const v4i g3 = {0, 0, 0, 0};
    const v8i g4 = {0, 0, 0, 0, 0, 0, 0, 0};              // VADDR4: unused -> zeros
    __builtin_amdgcn_tensor_load_to_lds(g0, g1, g2, g3, g4, 0);
}
#endif

__global__ __launch_bounds__(256) void wavg_color_kernel(
    const float* __restrict__ x_lab,
    const float* __restrict__ x_pred,
    float* __restrict__ out)
{
    __shared__ float sm[5 * HTILE];
    float* sm_l  = sm;
    float* sm_a  = sm + 1 * HTILE;
    float* sm_b  = sm + 2 * HTILE;
    float* sm_ap = sm + 3 * HTILE;
    float* sm_bp = sm + 4 * HTILE;

    const int tx  = threadIdx.x;             // 0..63
    const int ty  = threadIdx.y;             // 0..3
    const int tid = ty * TW + tx;            // 0..255
    const int x0  = blockIdx.x * TW;
    const int y0  = blockIdx.y * TH;
    const int n   = blockIdx.z;

    const float* gl  = x_lab  + (size_t)(3 * n + 0) * NPLANE;
    const float* ga  = x_lab  + (size_t)(3 * n + 1) * NPLANE;
    const float* gb  = x_lab  + (size_t)(3 * n + 2) * NPLANE;
    const float* gap = x_pred + (size_t)(3 * n + 1) * NPLANE;
    const float* gbp = x_pred + (size_t)(3 * n + 2) * NPLANE;

    // Low 32 bits of a generic pointer to __shared__ == wave-relative LDS addr.
    const uint32_t lds0 = (uint32_t)(size_t)(void*)sm;

#if HAVE_TDM
    const bool tdm_ok = (x0 > 0) && (y0 > 0);   // tile origin >= 0 in both dims
#else
    const bool tdm_ok = false;
#endif

    if (tdm_ok) {
#if HAVE_TDM
        // ---- Interior block: 5 TDM descriptors staged by wave 0 only -------
        if (tid < 32) {                       // whole-wave guard (TDM ignores EXEC)
            const int gx0 = x0 - 1, gy0 = y0 - 1;
            tdm_load_tile(gl,  lds0,                   gx0, gy0);
            tdm_load_tile(ga,  lds0 + 1u * HTILE * 4u, gx0, gy0);
            tdm_load_tile(gb,  lds0 + 2u * HTILE * 4u, gx0, gy0);
            tdm_load_tile(gap, lds0 + 3u * HTILE * 4u, gx0, gy0);
            tdm_load_tile(gbp, lds0 + 4u * HTILE * 4u, gx0, gy0);
            __builtin_amdgcn_s_wait_tensorcnt(0);
        }
#endif
    } else {
        // ---- Border block: per-lane async copies with explicit zero pad ----
        for (int idx = tid; idx < HTILE; idx += 256) {
            const int hy = idx / HTW;
            const int hx = idx - hy * HTW;
            const int gx = x0 - 1 + hx;
            const int gy = y0 - 1 + hy;
            const uint32_t lb = lds0 + (uint32_t)idx * 4u;
            if ((unsigned)gx < (unsigned)WW && (unsigned)gy < (unsigned)HH) {
                const size_t go = (size_t)gy * WW + gx;
                async_f32_to_lds(lb,                   gl  + go);
                async_f32_to_lds(lb + 1u * HTILE * 4u, ga  + go);
                async_f32_to_lds(lb + 2u * HTILE * 4u, gb  + go);
                async_f32_to_lds(lb + 3u * HTILE * 4u, gap + go);
                async_f32_to_lds(lb + 4u * HTILE * 4u, gbp + go);
            } else {
                // Zero padding: disjoint from async targets -> no DS/ASYNC hazard.
                sm_l[idx]  = 0.f; sm_a[idx]  = 0.f; sm_b[idx] = 0.f;
                sm_ap[idx] = 0.f; sm_bp[idx] = 0.f;
            }
        }
        asm volatile("s_wait_asynccnt 0" ::: "memory");
    }
    __syncthreads();   // block-uniform path choice -> barrier is uniform

    // ---- Compute: 4 output rows per thread ---------------------------------
#pragma unroll
    for (int r = 0; r < 4; ++r) {
        const int y = ty + 4 * r;                  // 0..15 within tile
        const int c = (y + 1) * HTW + (tx + 1);    // center in halo coords
        const float lc = sm_l[c];
        const float ac = sm_a[c];
        const float bc = sm_b[c];
        float s = 0.f, wa = 0.f, wb = 0.f;
#pragma unroll
        for (int dy = -1; dy <= 1; ++dy) {
#pragma unroll
            for (int dx = -1; dx <= 1; ++dx) {
                const int o = c + dy * HTW + dx;
                const float dl = sm_l[o] - lc;     // (50*dl)^2 = 2500*dl^2
                const float da = sm_a[o] - ac;
                const float db = sm_b[o] - bc;
                const float d  = fmaf(dl * 2500.f, dl, fmaf(da, da, db * db));
                const float e  = __expf(-d);       // d >= 0, max logit == 0
                s  += e;
                wa  = fmaf(e, sm_ap[o], wa);
                wb  = fmaf(e, sm_bp[o], wb);
            }
        }
        const float inv = __builtin_amdgcn_rcpf(s);
        const size_t row = (size_t)(y0 + y) * WW + (x0 + tx);
        out[(size_t)(2 * n + 0) * NPLANE + row] = wa * inv;
        out[(size_t)(2 * n + 1) * NPLANE + row] = wb * inv;
    }
}

extern "C" void kernel_launch(void* const* d_in, const int* in_sizes, int n_in,
                              void* d_out, int out_size, void* d_ws, size_t ws_size,
                              hipStream_t stream) {
    const float* x_lab  = (const float*)d_in[0];
    const float* x_pred = (const float*)d_in[1];
    float* out = (float*)d_out;

    const int N = in_sizes[0] / (3 * NPLANE);   // 8
    dim3 grid(WW / TW, HH / TH, N);             // (8, 32, 8) = 2048 blocks
    dim3 block(TW, 4, 1);                       // 256 threads = 8 wave32s
    hipLaunchKernelGGL(wavg_color_kernel, grid, block, 0, stream,
                       x_lab, x_pred, out);
}